// JKNet_5600637354059
// MI455X (gfx1250) — compile-verified
//
#include <hip/hip_runtime.h>
#include <hip/hip_bf16.h>

// ---------------------------------------------------------------------------
// JKNet (GCN x4 + JK-concat + linear) for gfx1250 (MI455X), wave32.
// Dense GEMMs use V_WMMA_F32_16X16X4_F32 (full fp32 precision) with:
//   - weights repacked once per layer into WMMA fragment order (one coalesced
//     global_load_b64 per lane per k-step, zero-padded columns -> no guards)
//   - A row-panel staged in LDS, shared by 8 column-tile waves per block,
//     row stride K+4 floats => conflict-free ds_load_b64 fragment reads
// ---------------------------------------------------------------------------

typedef __attribute__((ext_vector_type(2))) float v2f;
typedef __attribute__((ext_vector_type(8))) float v8f;

#define HDIM 128
#define LAYERS 4
#define BN_EPS 1e-5f

__device__ __forceinline__ void atomic_add_f32(float* p, float v) {
    __hip_atomic_fetch_add(p, v, __ATOMIC_RELAXED, __HIP_MEMORY_SCOPE_AGENT);
}

// ---------------- degree / norm --------------------------------------------

__global__ void deg_init_kernel(float* deg, int n) {
    int i = blockIdx.x * blockDim.x + threadIdx.x;
    if (i < n) deg[i] = 1.0f;               // self-loop
}

__global__ void deg_edges_kernel(float* deg, const int* dst, int e) {
    int i = blockIdx.x * blockDim.x + threadIdx.x;
    if (i < e) atomic_add_f32(&deg[dst[i]], 1.0f);
}

__global__ void rsqrt_kernel(float* deg, int n) {
    int i = blockIdx.x * blockDim.x + threadIdx.x;
    if (i < n) deg[i] = rsqrtf(deg[i]);     // deg -> dinv in place
}

// ---------------- weight repack into WMMA fragment order --------------------
// For column-tile nt, k-step k4 (k = 4*k4), lane l (half = l>>4, c = l&15):
//   Bp[((nt*K4 + k4)*32 + l)*2 + j] = B[(4*k4 + half*2 + j)*ldb + nt*16 + c]
// Columns >= Ncols are zero-padded (final GEMM: 40 -> 48).
__global__ void repack_B_kernel(const float* __restrict__ B,
                                float* __restrict__ Bp,
                                int K4, int NT, int Ncols, int ldb) {
    const int idx = blockIdx.x * blockDim.x + threadIdx.x;  // over NT*K4*32
    if (idx >= NT * K4 * 32) return;
    const int lane = idx & 31;
    const int t    = idx >> 5;
    const int k4   = t % K4;
    const int nt   = t / K4;
    const int col  = nt * 16 + (lane & 15);
    const int krow = k4 * 4 + (lane >> 4) * 2;
    float v0 = 0.0f, v1 = 0.0f;
    if (col < Ncols) {
        v0 = B[(long long)krow * ldb + col];
        v1 = B[(long long)(krow + 1) * ldb + col];
    }
    Bp[(size_t)idx * 2 + 0] = v0;
    Bp[(size_t)idx * 2 + 1] = v1;
}

// ---------------- fp32 WMMA GEMM -------------------------------------------
// grid.x = M/16 row tiles; blockDim = 32*NT (wave w owns column tile w).
// A panel (16 rows x K) cooperatively staged in LDS (stride K+4 floats:
// K multiple of 64 => 4-dword bank step per lane => conflict-free b64 reads).
// Fragment layouts per CDNA5 ISA (32-bit A 16x4 / B 4x16 / C 16x16):
//   A: lane(0..15) holds row M=lane&15, v2f = A[row][k + (lane>>4)*2 + {0,1}]
//   B: lane(0..15) holds col N=lane&15 (from repacked Bp, linear reads)
//   C: VGPR r -> M = (lane>>4)*8 + r, N = lane&15
__global__ __launch_bounds__(256)
void gemm_wmma_f32(const float* __restrict__ A,
                   const float* __restrict__ Bp,
                   float* __restrict__ C,
                   int K, int kshift /* log2(K) */, int Ncols,
                   int lda, int ldc,
                   const float* __restrict__ bias) {
    extern __shared__ float sA[];           // 16 * (K+4) floats
    const int mt   = blockIdx.x;
    const int tid  = threadIdx.x;
    const int lane = tid & 31;
    const int wave = tid >> 5;              // column tile index
    const int half = lane >> 4;
    const int l15  = lane & 15;
    const int K4   = K >> 2;
    const int sstr = K + 4;                 // LDS row stride (floats)

    // cooperative stage of the 16-row A panel (coalesced along rows)
    const long long abase = (long long)(mt * 16) * lda;
    for (int idx = tid; idx < 16 * K; idx += blockDim.x) {
        const int r = idx >> kshift;
        const int k = idx & (K - 1);
        sA[r * sstr + k] = A[abase + (long long)r * lda + k];
    }
    __syncthreads();

    const float* ap = &sA[l15 * sstr + half * 2];
    const float* bp = Bp + ((size_t)wave * K4 * 32 + lane) * 2;

    v8f acc = {};
    for (int k4 = 0; k4 < K4; ++k4) {
        const v2f a = *(const v2f*)(ap + k4 * 4);        // ds_load_b64
        const v2f b = *(const v2f*)(bp + (size_t)k4 * 64); // global_load_b64
        acc = __builtin_amdgcn_wmma_f32_16x16x4_f32(
            /*neg_a=*/false, a, /*neg_b=*/false, b,
            /*c_mod=*/(short)0, acc, /*reuse_a=*/false, /*reuse_b=*/false);
    }

    const int ncol = wave * 16 + l15;
    if (ncol < Ncols) {
        const float bv = bias ? bias[ncol] : 0.0f;
        #pragma unroll
        for (int r = 0; r < 8; ++r) {
            const int m = mt * 16 + half * 8 + r;
            C[(long long)m * ldc + ncol] = acc[r] + bv;
        }
    }
}

// ---------------- aggregation ----------------------------------------------

// agg[i][f] = hw[i][f] * dinv[i]^2 + bias[f]   (self-loop term + bias)
__global__ void agg_init_kernel(const float* __restrict__ hw,
                                const float* __restrict__ dinv,
                                const float* __restrict__ bias,
                                float* __restrict__ agg, int n) {
    long long idx = (long long)blockIdx.x * blockDim.x + threadIdx.x;
    if (idx >= (long long)n * HDIM) return;
    const int row = (int)(idx >> 7);
    const int f   = (int)(idx & (HDIM - 1));
    const float dv = dinv[row];
    agg[idx] = hw[idx] * dv * dv + bias[f];
}

// 32 lanes per edge, 4 floats per lane (128 features).
// agg[dst] += hw[src] * dinv[src]*dinv[dst]  (hardware global_atomic_add_f32)
__global__ void agg_edges_kernel(const float* __restrict__ hw,
                                 const int* __restrict__ src,
                                 const int* __restrict__ dst,
                                 const float* __restrict__ dinv,
                                 float* __restrict__ agg, int e) {
    long long gid = (long long)blockIdx.x * blockDim.x + threadIdx.x;
    const int edge = (int)(gid >> 5);
    if (edge >= e) return;
    const int lane = (int)(gid & 31);
    const int s = src[edge];
    const int d = dst[edge];
    const float nrm = dinv[s] * dinv[d];
    const float4 v = ((const float4*)(hw + (long long)s * HDIM))[lane];
    float* arow = agg + (long long)d * HDIM + lane * 4;
    atomic_add_f32(arow + 0, v.x * nrm);
    atomic_add_f32(arow + 1, v.y * nrm);
    atomic_add_f32(arow + 2, v.z * nrm);
    atomic_add_f32(arow + 3, v.w * nrm);
}

// ---------------- batch-norm -----------------------------------------------

__global__ void zero256_kernel(float* p) { p[threadIdx.x] = 0.0f; }

// blockDim = 128 (one thread per feature), grid-strided over rows.
__global__ void bn_stats_kernel(const float* __restrict__ agg,
                                float* __restrict__ sums, int n) {
    const int f = threadIdx.x;
    float s = 0.0f, sq = 0.0f;
    for (int r = blockIdx.x; r < n; r += gridDim.x) {
        const float v = agg[(long long)r * HDIM + f];
        s += v;
        sq += v * v;
    }
    atomic_add_f32(&sums[f], s);
    atomic_add_f32(&sums[HDIM + f], sq);
}

__global__ void bn_finalize_kernel(const float* __restrict__ sums,
                                   const float* __restrict__ gamma,
                                   const float* __restrict__ beta,
                                   float* __restrict__ scsh, int n) {
    const int f = threadIdx.x;
    const float inv_n = 1.0f / (float)n;
    const float mu  = sums[f] * inv_n;
    const float var = sums[HDIM + f] * inv_n - mu * mu;   // population var
    const float sc  = gamma[f] * rsqrtf(var + BN_EPS);
    scsh[f]        = sc;
    scsh[HDIM + f] = beta[f] - mu * sc;
}

// h = relu(agg*scale + shift), written into hcat[:, layer*128 : (layer+1)*128]
__global__ void bn_apply_kernel(const float* __restrict__ agg,
                                const float* __restrict__ scsh,
                                float* __restrict__ hcat,
                                int layer, int n) {
    long long idx = (long long)blockIdx.x * blockDim.x + threadIdx.x;
    if (idx >= (long long)n * HDIM) return;
    const int row = (int)(idx >> 7);
    const int f   = (int)(idx & (HDIM - 1));
    const float v = agg[idx] * scsh[f] + scsh[HDIM + f];
    hcat[(long long)row * (HDIM * LAYERS) + layer * HDIM + f] = fmaxf(v, 0.0f);
}

// ---------------- host-side orchestration ----------------------------------

extern "C" void kernel_launch(void* const* d_in, const int* in_sizes, int n_in,
                              void* d_out, int out_size, void* d_ws, size_t ws_size,
                              hipStream_t stream) {
    const float* x      = (const float*)d_in[0];   // [N, 256]
    const int*   ei     = (const int*)  d_in[1];   // [2, E] (src row, dst row)
    const float* W0     = (const float*)d_in[2];   // [256, 128]
    const float* b0     = (const float*)d_in[3];   // [128]
    const float* Ws     = (const float*)d_in[4];   // [3, 128, 128]
    const float* bs     = (const float*)d_in[5];   // [3, 128]
    const float* gammas = (const float*)d_in[6];   // [4, 128]
    const float* betas  = (const float*)d_in[7];   // [4, 128]
    const float* Wlin   = (const float*)d_in[8];   // [512, 40]
    const float* blin   = (const float*)d_in[9];   // [40]
    float* out = (float*)d_out;                    // [N, 40]

    const int F_IN = 256;
    const int C    = 40;
    const int N = in_sizes[0] / F_IN;
    const int E = in_sizes[1] / 2;
    const int* src = ei;
    const int* dst = ei + E;

    // workspace carve-up (256B aligned)
    auto align_up = [](size_t v) { return (v + 255) & ~(size_t)255; };
    char* ws = (char*)d_ws;
    size_t off = 0;
    float* dinv = (float*)(ws + off); off = align_up(off + (size_t)N * 4);
    float* hw   = (float*)(ws + off); off = align_up(off + (size_t)N * HDIM * 4);
    float* agg  = (float*)(ws + off); off = align_up(off + (size_t)N * HDIM * 4);
    float* hcat = (float*)(ws + off); off = align_up(off + (size_t)N * HDIM * LAYERS * 4);
    float* sums = (float*)(ws + off); off = align_up(off + 256 * 4);
    float* scsh = (float*)(ws + off); off = align_up(off + 256 * 4);
    float* Bp   = (float*)(ws + off); off = align_up(off + (size_t)32768 * 4); // max repacked W
    (void)ws_size; (void)n_in; (void)out_size;

    const int TB = 256;
    const long long nh = (long long)N * HDIM;

    // --- GCN symmetric norm: dinv = rsqrt(1 + in_degree) --------------------
    deg_init_kernel <<<(N + TB - 1) / TB, TB, 0, stream>>>(dinv, N);
    deg_edges_kernel<<<(E + TB - 1) / TB, TB, 0, stream>>>(dinv, dst, E);
    rsqrt_kernel    <<<(N + TB - 1) / TB, TB, 0, stream>>>(dinv, N);

    // --- GCN layers ---------------------------------------------------------
    for (int layer = 0; layer < LAYERS; ++layer) {
        const float* A;
        int K, kshift, lda;
        if (layer == 0) { A = x; K = F_IN; kshift = 8; lda = F_IN; }
        else { A = hcat + (size_t)(layer - 1) * HDIM; K = HDIM; kshift = 7; lda = HDIM * LAYERS; }
        const float* W = (layer == 0) ? W0 : (Ws + (size_t)(layer - 1) * HDIM * HDIM);
        const float* b = (layer == 0) ? b0 : (bs + (size_t)(layer - 1) * HDIM);
        const int NT = HDIM / 16;           // 8 column tiles
        const int K4 = K / 4;

        // repack W into fragment order, then hw = A @ W (fp32 WMMA)
        repack_B_kernel<<<(NT * K4 * 32 + TB - 1) / TB, TB, 0, stream>>>(
            W, Bp, K4, NT, HDIM, HDIM);
        gemm_wmma_f32<<<N / 16, 32 * NT, 16 * (K + 4) * 4, stream>>>(
            A, Bp, hw, K, kshift, HDIM, lda, HDIM, nullptr);

        // agg = self-loop + bias, then scatter-add edge messages
        agg_init_kernel<<<(int)((nh + TB - 1) / TB), TB, 0, stream>>>(hw, dinv, b, agg, N);
        agg_edges_kernel<<<(int)(((long long)E * 32 + TB - 1) / TB), TB, 0, stream>>>(
            hw, src, dst, dinv, agg, E);

        // batch-norm (training-mode batch stats) + relu, into hcat slice
        zero256_kernel<<<1, 256, 0, stream>>>(sums);
        bn_stats_kernel<<<1024, HDIM, 0, stream>>>(agg, sums, N);
        bn_finalize_kernel<<<1, HDIM, 0, stream>>>(
            sums, gammas + (size_t)layer * HDIM, betas + (size_t)layer * HDIM, scsh, N);
        bn_apply_kernel<<<(int)((nh + TB - 1) / TB), TB, 0, stream>>>(
            agg, scsh, hcat, layer, N);
    }

    // --- final linear: out = hcat @ Wlin + blin -----------------------------
    {
        const int K = HDIM * LAYERS;        // 512
        const int NT = (C + 15) / 16;       // 3 tiles (48 cols, zero-padded)
        const int K4 = K / 4;
        repack_B_kernel<<<(NT * K4 * 32 + TB - 1) / TB, TB, 0, stream>>>(
            Wlin, Bp, K4, NT, C, C);
        gemm_wmma_f32<<<N / 16, 32 * NT, 16 * (K + 4) * 4, stream>>>(
            hcat, Bp, out, K, 9, C, K, C, blin);
    }
}